// GraphConvDQL_72241349918781
// MI455X (gfx1250) — compile-verified
//
#include <hip/hip_runtime.h>
#include <math.h>

typedef float v2f __attribute__((ext_vector_type(2)));
typedef float v8f __attribute__((ext_vector_type(8)));

#define STRIDE 12   // padded row stride for 10-channel rows -> 48B, 16B-aligned

// -------------------------------------------------------------------------
// Kernel A: y[n,0:10] = x[n,0:16] @ Wrel1^T  via V_WMMA_F32_16X16X4_F32.
// One wave handles a 16-node tile: A = 16x16 x-tile consumed as four 16x4
// slabs, B = Wrel^T consumed as four 4x16 slabs, C accumulates over K.
// Also zeroes agg rows for this tile (must happen before the scatter pass).
// -------------------------------------------------------------------------
__global__ void k_node_y_wmma(const float* __restrict__ x,
                              const float* __restrict__ Wrel,
                              float* __restrict__ y,
                              float* __restrict__ agg) {
  const int wave = threadIdx.x >> 5;
  const int lane = threadIdx.x & 31;
  const long long tile = (long long)blockIdx.x * 8 + wave;  // 16 nodes / tile
  const long long base = tile * 16;
  const int m  = lane & 15;   // row (A) / column (B,C,D) index held by lane
  const int hi = lane >> 4;   // lane half selects K (A/B) or M-half (C/D)

  // A operand slabs: lane holds X[base+m][k0+2*hi .. k0+2*hi+1]
  const float* xrow = x + (base + m) * 16;
  v2f a0 = *(const v2f*)(xrow + 0  + 2 * hi);
  v2f a1 = *(const v2f*)(xrow + 4  + 2 * hi);
  v2f a2 = *(const v2f*)(xrow + 8  + 2 * hi);
  v2f a3 = *(const v2f*)(xrow + 12 + 2 * hi);

  // B operand slabs: B[k][n] = Wrel[n][k]; columns n>=10 are zero.
  v2f b0 = {0.f, 0.f}, b1 = {0.f, 0.f}, b2 = {0.f, 0.f}, b3 = {0.f, 0.f};
  if (m < 10) {
    const float* wrow = Wrel + m * 16;
    b0 = *(const v2f*)(wrow + 0  + 2 * hi);
    b1 = *(const v2f*)(wrow + 4  + 2 * hi);
    b2 = *(const v2f*)(wrow + 8  + 2 * hi);
    b3 = *(const v2f*)(wrow + 12 + 2 * hi);
  }

  v8f c = {0.f, 0.f, 0.f, 0.f, 0.f, 0.f, 0.f, 0.f};
  c = __builtin_amdgcn_wmma_f32_16x16x4_f32(false, a0, false, b0, (short)0, c, false, false);
  c = __builtin_amdgcn_wmma_f32_16x16x4_f32(false, a1, false, b1, (short)0, c, false, false);
  c = __builtin_amdgcn_wmma_f32_16x16x4_f32(false, a2, false, b2, (short)0, c, false, false);
  c = __builtin_amdgcn_wmma_f32_16x16x4_f32(false, a3, false, b3, (short)0, c, false, false);

  // D layout: VGPR r, lanes 0-15 -> row r col m ; lanes 16-31 -> row r+8.
  if (m < 10) {
    const long long nb = base + hi * 8;
#pragma unroll
    for (int r = 0; r < 8; ++r)
      y[(nb + r) * STRIDE + m] = c[r];
  }

  // zero agg for this tile: 16*12 = 192 floats, coalesced
  float* ag = agg + base * STRIDE;
#pragma unroll
  for (int i = 0; i < 6; ++i) ag[lane + i * 32] = 0.f;
}

// -------------------------------------------------------------------------
// Kernel B: per-graph [start,end) from sorted batch ids (== bincount+cumsum
// of the reference for sorted segment ids), plus zero the stat bins.
// -------------------------------------------------------------------------
__global__ void k_bounds(const int* __restrict__ batch, int N,
                         int* __restrict__ startA, int* __restrict__ endA,
                         float* __restrict__ statZero, int nz) {
  const int n = blockIdx.x * blockDim.x + threadIdx.x;
  if (n < nz) statZero[n] = 0.f;
  if (n >= N) return;
  const int b = batch[n];
  if (n == 0 || batch[n - 1] != b) startA[b] = n;
  if (n == N - 1 || batch[n + 1] != b) endA[b] = n + 1;
}

// -------------------------------------------------------------------------
// Kernel C: edge scatter  agg[dst] += ew * y[src]   (10 channels)
// y and agg are L2-resident (N*48B each << 192MB L2).
// -------------------------------------------------------------------------
__global__ void k_scatter(const int* __restrict__ ei, const float* __restrict__ ew,
                          const float* __restrict__ y, float* __restrict__ agg, int E) {
  const int e = blockIdx.x * blockDim.x + threadIdx.x;
  if (e >= E) return;
  const int s = ei[e];
  const int d = ei[E + e];
  const float w = ew[e];
  const float4* yr = (const float4*)(y + (long long)s * STRIDE);
  const float4 p0 = yr[0], p1 = yr[1], p2 = yr[2];
  float* ag = agg + (long long)d * STRIDE;
  atomicAdd(ag + 0, w * p0.x); atomicAdd(ag + 1, w * p0.y);
  atomicAdd(ag + 2, w * p0.z); atomicAdd(ag + 3, w * p0.w);
  atomicAdd(ag + 4, w * p1.x); atomicAdd(ag + 5, w * p1.y);
  atomicAdd(ag + 6, w * p1.z); atomicAdd(ag + 7, w * p1.w);
  atomicAdd(ag + 8, w * p2.x); atomicAdd(ag + 9, w * p2.y);
}

// -------------------------------------------------------------------------
// Kernel D: per-graph Sum(h) and Sum(h^2), h = agg + x@Wroot^T + brel.
// WMMA version: one wave computes r = X_tile @ Wroot^T (+brel via the
// initial C operand) for a 16-node tile; the C/D layout puts 8 nodes of
// channel c in the v8f of the lane holding column c, so per-channel Sum(h)
// and Sum(h^2) reduce entirely within the lane, then 2 LDS atomics/lane.
// Block = 256 threads = 8 waves = 128 nodes (NPG divisible by 128).
// -------------------------------------------------------------------------
__global__ void k_reduce_wmma(const float* __restrict__ agg, const float* __restrict__ x,
                              const float* __restrict__ Wroot, const float* __restrict__ brel,
                              float* __restrict__ sums, float* __restrict__ sumsq, int NPG) {
  __shared__ float lsum[20];
  const int tid  = threadIdx.x;
  const int wave = tid >> 5;
  const int lane = tid & 31;
  const int m  = lane & 15;
  const int hi = lane >> 4;

  if (tid < 20) lsum[tid] = 0.f;
  __syncthreads();

  const long long blockBase = (long long)blockIdx.x * 128;
  const int g = (int)(blockBase / NPG);
  const long long base = blockBase + (long long)wave * 16;   // this wave's 16 nodes

  // A operand slabs: X[base+m][k0+2*hi .. +1]
  const float* xrow = x + (base + m) * 16;
  v2f a0 = *(const v2f*)(xrow + 0  + 2 * hi);
  v2f a1 = *(const v2f*)(xrow + 4  + 2 * hi);
  v2f a2 = *(const v2f*)(xrow + 8  + 2 * hi);
  v2f a3 = *(const v2f*)(xrow + 12 + 2 * hi);

  // B operand slabs: B[k][n] = Wroot[n][k]; columns n>=10 zero.
  v2f b0 = {0.f, 0.f}, b1 = {0.f, 0.f}, b2 = {0.f, 0.f}, b3 = {0.f, 0.f};
  float bias = 0.f;
  if (m < 10) {
    const float* wrow = Wroot + m * 16;
    b0 = *(const v2f*)(wrow + 0  + 2 * hi);
    b1 = *(const v2f*)(wrow + 4  + 2 * hi);
    b2 = *(const v2f*)(wrow + 8  + 2 * hi);
    b3 = *(const v2f*)(wrow + 12 + 2 * hi);
    bias = brel[m];
  }

  // C initialized with the bias: D[r][c] starts at brel[c] for every row.
  v8f c = {bias, bias, bias, bias, bias, bias, bias, bias};
  c = __builtin_amdgcn_wmma_f32_16x16x4_f32(false, a0, false, b0, (short)0, c, false, false);
  c = __builtin_amdgcn_wmma_f32_16x16x4_f32(false, a1, false, b1, (short)0, c, false, false);
  c = __builtin_amdgcn_wmma_f32_16x16x4_f32(false, a2, false, b2, (short)0, c, false, false);
  c = __builtin_amdgcn_wmma_f32_16x16x4_f32(false, a3, false, b3, (short)0, c, false, false);

  if (m < 10) {
    // lane owns channel m for nodes base+hi*8 .. +7
    const float* ar = agg + (base + hi * 8) * STRIDE + m;
    float s1 = 0.f, s2 = 0.f;
#pragma unroll
    for (int r = 0; r < 8; ++r) {
      const float h = c[r] + ar[(long long)r * STRIDE];
      s1 += h;
      s2 += h * h;
    }
    atomicAdd(&lsum[m], s1);
    atomicAdd(&lsum[10 + m], s2);
  }
  __syncthreads();
  if (tid < 10) atomicAdd(&sums[g * 10 + tid], lsum[tid]);
  else if (tid < 20) atomicAdd(&sumsq[g * 10 + (tid - 10)], lsum[tid]);
}

// -------------------------------------------------------------------------
// Kernel E: head. One thread per graph: GraphNorm at gidx, adv/val MLPs,
// dueling combine, action mask.
// -------------------------------------------------------------------------
__global__ void k_head(const float* __restrict__ x, const int* __restrict__ cur,
                       const int* __restrict__ mask, const float* __restrict__ goal,
                       const float* __restrict__ agg,
                       const float* __restrict__ Wroot, const float* __restrict__ brel,
                       const float* __restrict__ gnw, const float* __restrict__ gnb,
                       const float* __restrict__ gnms,
                       const float* __restrict__ Wa1, const float* __restrict__ ba1,
                       const float* __restrict__ Wa2, const float* __restrict__ ba2,
                       const float* __restrict__ Wv1, const float* __restrict__ bv1,
                       const float* __restrict__ Wv2, const float* __restrict__ bv2,
                       const float* __restrict__ sums, const float* __restrict__ sumsq,
                       const int* __restrict__ startA, const int* __restrict__ endA,
                       float* __restrict__ out, int B) {
  const int b = blockIdx.x * blockDim.x + threadIdx.x;
  if (b >= B) return;
  const int st = startA[b];
  const float inv = 1.f / (float)(endA[b] - st);
  const long long gi = (long long)st + cur[b];
  const float* xr = x + gi * 16;
  const float* ar = agg + gi * STRIDE;

  float xcat[18];
#pragma unroll
  for (int c = 0; c < 10; ++c) {
    float h = ar[c] + brel[c];
#pragma unroll
    for (int k = 0; k < 16; ++k) h += xr[k] * Wroot[c * 16 + k];
    const float mean = sums[b * 10 + c] * inv;
    const float e2   = sumsq[b * 10 + c] * inv;
    const float ms   = gnms[c];
    const float var  = e2 - mean * mean * ms * (2.f - ms);
    const float o    = h - mean * ms;
    const float s    = gnw[c] * o * rsqrtf(var + 1e-5f) + gnb[c];
    xcat[c] = s > 0.f ? s : 0.f;
  }
#pragma unroll
  for (int k = 0; k < 8; ++k) xcat[10 + k] = goal[b * 8 + k];

  float ha[10], hv[10];
#pragma unroll
  for (int j = 0; j < 10; ++j) {
    float ta = ba1[j], tv = bv1[j];
#pragma unroll
    for (int k = 0; k < 18; ++k) {
      ta += xcat[k] * Wa1[j * 18 + k];
      tv += xcat[k] * Wv1[j * 18 + k];
    }
    ha[j] = ta > 0.f ? ta : 0.f;
    hv[j] = tv > 0.f ? tv : 0.f;
  }
  float val = bv2[0];
#pragma unroll
  for (int j = 0; j < 10; ++j) val += hv[j] * Wv2[j];

  float adv[50];
  float asum = 0.f;
#pragma unroll
  for (int a = 0; a < 50; ++a) {
    float t = ba2[a];
#pragma unroll
    for (int j = 0; j < 10; ++j) t += ha[j] * Wa2[a * 10 + j];
    adv[a] = t;
    asum += t;
  }
  const float am = asum / 50.0f;
#pragma unroll
  for (int a = 0; a < 50; ++a) {
    const float q = val + adv[a] - am;
    out[b * 50 + a] = (mask[b * 50 + a] == 0) ? -100000000.0f : q;
  }
}

// -------------------------------------------------------------------------
extern "C" void kernel_launch(void* const* d_in, const int* in_sizes, int n_in,
                              void* d_out, int out_size, void* d_ws, size_t ws_size,
                              hipStream_t stream) {
  const float* x     = (const float*)d_in[0];
  const int*   ei    = (const int*)d_in[1];
  const float* ew    = (const float*)d_in[2];
  const int*   batch = (const int*)d_in[3];
  const int*   cur   = (const int*)d_in[4];
  const int*   amask = (const int*)d_in[5];
  const float* goal  = (const float*)d_in[6];
  const float* Wrel1 = (const float*)d_in[7];
  const float* brel1 = (const float*)d_in[8];
  const float* Wroot1= (const float*)d_in[9];
  const float* gn1w  = (const float*)d_in[10];
  const float* gn1b  = (const float*)d_in[11];
  const float* gn1ms = (const float*)d_in[12];
  // d_in[13..18]: layer-2 params -> dead code in the reference forward
  const float* Wa1 = (const float*)d_in[21];
  const float* ba1 = (const float*)d_in[22];
  const float* Wa2 = (const float*)d_in[23];
  const float* ba2 = (const float*)d_in[24];
  const float* Wv1 = (const float*)d_in[25];
  const float* bv1 = (const float*)d_in[26];
  const float* Wv2 = (const float*)d_in[27];
  const float* bv2 = (const float*)d_in[28];

  const int N   = in_sizes[0] / 16;
  const int E   = in_sizes[2];
  const int B   = in_sizes[4];
  const int NPG = N / B;

  float* y     = (float*)d_ws;                 // N*12 f32
  float* agg   = y + (size_t)N * STRIDE;       // N*12 f32
  float* sums  = agg + (size_t)N * STRIDE;     // B*10 f32
  float* sumsq = sums + (size_t)B * 10;        // B*10 f32
  int*   startA = (int*)(sumsq + (size_t)B * 10);
  int*   endA   = startA + B;

  k_node_y_wmma<<<N / 128, 256, 0, stream>>>(x, Wrel1, y, agg);
  k_bounds<<<(N + 255) / 256, 256, 0, stream>>>(batch, N, startA, endA, sums, B * 20);
  k_scatter<<<(E + 255) / 256, 256, 0, stream>>>(ei, ew, y, agg, E);
  k_reduce_wmma<<<N / 128, 256, 0, stream>>>(agg, x, Wroot1, brel1, sums, sumsq, NPG);
  k_head<<<(B + 127) / 128, 128, 0, stream>>>(x, cur, amask, goal, agg, Wroot1, brel1,
                                              gn1w, gn1b, gn1ms, Wa1, ba1, Wa2, ba2,
                                              Wv1, bv1, Wv2, bv2, sums, sumsq,
                                              startA, endA, (float*)d_out, B);
}